// MixProp_29764123361444
// MI455X (gfx1250) — compile-verified
//
#include <hip/hip_runtime.h>

// ---------------------------------------------------------------------------
// MixProp on MI455X (gfx1250, wave32, WMMA).
//
//   y = b + E0@x + E1@g1 + E2@g2 + E3@g3,  g_{k+1} = Anorm @ g_k
// where Anorm = rownorm(adj + I) and E_k fold the alpha-residual mix into
// the 1x1-conv weights (exact algebraic refactor of the reference).
//
// Memory-bound (~4.6 GB moved vs 224 GFLOP) -> fp32 WMMA 16x16x4 everywhere.
// ---------------------------------------------------------------------------

typedef __attribute__((ext_vector_type(2))) float v2f;
typedef __attribute__((ext_vector_type(8))) float v8f;

#define NV 512                       // nodes (v) == sequence (l)
#define NB 8                         // batch
#define CH 32                        // channels
#define PIX ((size_t)NV * NV)        // pixels per (n, channel) slab = 262144
#define SLABF ((size_t)NB * CH * PIX) // floats per full tensor

// ---------------------------------------------------------------------------
// Kernel 1: Anorm[v][w] = (adj[v][w] + (v==w)) / rowsum(adj[v] + I)
// ---------------------------------------------------------------------------
__global__ __launch_bounds__(128) void norm_adj_kernel(const float* __restrict__ adj,
                                                       float* __restrict__ An) {
    __shared__ float red[128];
    const int row = blockIdx.x;
    const int tid = threadIdx.x;
    float s = 0.0f;
    for (int c = tid; c < NV; c += 128)
        s += adj[row * NV + c] + (c == row ? 1.0f : 0.0f);
    red[tid] = s;
    __syncthreads();
    for (int off = 64; off > 0; off >>= 1) {
        if (tid < off) red[tid] += red[tid + off];
        __syncthreads();
    }
    const float inv = 1.0f / red[0];
    for (int c = tid; c < NV; c += 128)
        An[row * NV + c] = (adj[row * NV + c] + (c == row ? 1.0f : 0.0f)) * inv;
}

// ---------------------------------------------------------------------------
// Kernel 2: effective conv weights E0..E3 [4][32][32] from W [32][128]
//   E0 = W0 + a(W1+W2+W3); E1 = (1-a)(W1 + aW2 + aW3)
//   E2 = (1-a)^2 (W2 + aW3); E3 = (1-a)^3 W3
// ---------------------------------------------------------------------------
__global__ __launch_bounds__(256) void make_eff_w_kernel(const float* __restrict__ W,
                                                         float* __restrict__ E) {
    const int i = blockIdx.x * 256 + threadIdx.x;   // 0..1023  (o*32 + c)
    if (i >= CH * CH) return;
    const int o = i >> 5, c = i & 31;
    const float a = 0.05f, om = 0.95f;
    const float w0 = W[o * 128 + c];
    const float w1 = W[o * 128 + 32 + c];
    const float w2 = W[o * 128 + 64 + c];
    const float w3 = W[o * 128 + 96 + c];
    E[0 * 1024 + i] = w0 + a * (w1 + w2 + w3);
    E[1 * 1024 + i] = om * (w1 + a * w2 + a * w3);
    E[2 * 1024 + i] = om * om * (w2 + a * w3);
    E[3 * 1024 + i] = om * om * om * w3;
}

// ---------------------------------------------------------------------------
// Kernel 3: diffusion GEMM, per slab s in [0,256): Gout[s] = Anorm @ Gin[s]
//   M=N=K=512. Workgroup: 64x64 tile, 8 waves (4 M-strips x 2 N-groups),
//   each wave owns one 16-row strip and two 16x16 WMMA accumulators.
//   LDS stage: A 64x32 (stride 36), B 32x64 (stride 68) per K-step.
// ---------------------------------------------------------------------------
__global__ __launch_bounds__(256) void diffuse_kernel(const float* __restrict__ Gin,
                                                      float* __restrict__ Gout,
                                                      const float* __restrict__ A) {
    __shared__ float As[64][36];   // stride 36: 16B-aligned rows, conflict-free frag reads
    __shared__ float Bs[32][68];   // stride 68: 16B-aligned rows

    const int slab = blockIdx.z;
    const int m0 = blockIdx.y * 64;
    const int n0 = blockIdx.x * 64;
    const float* gin  = Gin  + (size_t)slab * PIX;
    float*       gout = Gout + (size_t)slab * PIX;

    const int tid   = threadIdx.x;
    const int wave  = tid >> 5;
    const int lane  = tid & 31;
    const int lm    = lane & 15;      // m (A frag) / n (B,C,D frags) within tile
    const int kh    = lane >> 4;      // k-half selector (K=0,1 vs K=2,3)
    const int waveM = wave & 3;       // 4 strips of 16 rows
    const int waveN = wave >> 2;      // 2 groups of 32 cols

    v8f acc0 = {};
    v8f acc1 = {};

    for (int k0 = 0; k0 < NV; k0 += 32) {
        __syncthreads();
        // A tile: 64 rows x 32 k = 512 float4, 2 per thread
        #pragma unroll
        for (int it = 0; it < 2; ++it) {
            const int f = tid + it * 256;
            const int row = f >> 3, col = (f & 7) * 4;
            const float4 v = *(const float4*)&A[(size_t)(m0 + row) * NV + k0 + col];
            *(float4*)&As[row][col] = v;
        }
        // B tile: 32 k x 64 cols = 512 float4, 2 per thread
        #pragma unroll
        for (int it = 0; it < 2; ++it) {
            const int f = tid + it * 256;
            const int row = f >> 4, col = (f & 15) * 4;
            const float4 v = *(const float4*)&gin[(size_t)(k0 + row) * NV + n0 + col];
            *(float4*)&Bs[row][col] = v;
        }
        __syncthreads();

        #pragma unroll
        for (int kk = 0; kk < 32; kk += 4) {
            v2f af;
            af.x = As[waveM * 16 + lm][kk + 2 * kh];
            af.y = As[waveM * 16 + lm][kk + 2 * kh + 1];
            v2f b0;
            b0.x = Bs[kk + 2 * kh][waveN * 32 + lm];
            b0.y = Bs[kk + 2 * kh + 1][waveN * 32 + lm];
            v2f b1;
            b1.x = Bs[kk + 2 * kh][waveN * 32 + 16 + lm];
            b1.y = Bs[kk + 2 * kh + 1][waveN * 32 + 16 + lm];
            acc0 = __builtin_amdgcn_wmma_f32_16x16x4_f32(false, af, false, b0,
                                                         (short)0, acc0, false, false);
            acc1 = __builtin_amdgcn_wmma_f32_16x16x4_f32(false, af, false, b1,
                                                         (short)0, acc1, false, false);
        }
    }

    #pragma unroll
    for (int r = 0; r < 8; ++r) {
        const int row = m0 + waveM * 16 + r + 8 * kh;
        gout[(size_t)row * NV + n0 + waveN * 32 + lm]      = acc0[r];
        gout[(size_t)row * NV + n0 + waveN * 32 + 16 + lm] = acc1[r];
    }
}

// ---------------------------------------------------------------------------
// Kernel 4: channel-mix GEMM, per batch n:  Y[n,o,p] (init/accum)= Ek @ G[n,c,p]
//   M=32 (o), K=32 (c), N=128 pixels per workgroup. 8 waves: 2 o-strips x
//   4 pixel groups, each wave 2 WMMA tiles. Acc seeded with bias (pass 0)
//   or with existing Y (passes 1..3).
// ---------------------------------------------------------------------------
__global__ __launch_bounds__(256) void conv_kernel(const float* __restrict__ G,
                                                   const float* __restrict__ E,
                                                   const float* __restrict__ bias,
                                                   float* __restrict__ Y, int init) {
    __shared__ float Es[32][33];    // E[o][c], stride 33 -> conflict-free column reads
    __shared__ float Gs[32][132];   // G tile [c][p], stride 132 (16B-aligned rows)

    const int n  = blockIdx.y;
    const int p0 = blockIdx.x * 128;
    const float* g = G + (size_t)n * CH * PIX;
    float*       y = Y + (size_t)n * CH * PIX;
    const int tid = threadIdx.x;

    for (int i = tid; i < CH * CH; i += 256) Es[i >> 5][i & 31] = E[i];
    #pragma unroll
    for (int it = 0; it < 4; ++it) {
        const int f = tid + it * 256;           // 1024 float4 total
        const int row = f >> 5, col = (f & 31) * 4;
        const float4 v = *(const float4*)&g[(size_t)row * PIX + p0 + col];
        *(float4*)&Gs[row][col] = v;
    }
    __syncthreads();

    const int wave = tid >> 5, lane = tid & 31;
    const int lm = lane & 15, kh = lane >> 4;
    const int mt = (wave & 1) * 16;     // o strip base
    const int nb = (wave >> 1) * 32;    // pixel group base within 128

    v8f acc0 = {};
    v8f acc1 = {};
    #pragma unroll
    for (int r = 0; r < 8; ++r) {
        const int orow = mt + r + 8 * kh;
        if (init) {
            const float bv = bias[orow];
            acc0[r] = bv;
            acc1[r] = bv;
        } else {
            acc0[r] = y[(size_t)orow * PIX + p0 + nb + lm];
            acc1[r] = y[(size_t)orow * PIX + p0 + nb + 16 + lm];
        }
    }

    #pragma unroll
    for (int kk = 0; kk < 32; kk += 4) {
        v2f af;
        af.x = Es[mt + lm][kk + 2 * kh];
        af.y = Es[mt + lm][kk + 2 * kh + 1];
        v2f b0;
        b0.x = Gs[kk + 2 * kh][nb + lm];
        b0.y = Gs[kk + 2 * kh + 1][nb + lm];
        v2f b1;
        b1.x = Gs[kk + 2 * kh][nb + 16 + lm];
        b1.y = Gs[kk + 2 * kh + 1][nb + 16 + lm];
        acc0 = __builtin_amdgcn_wmma_f32_16x16x4_f32(false, af, false, b0,
                                                     (short)0, acc0, false, false);
        acc1 = __builtin_amdgcn_wmma_f32_16x16x4_f32(false, af, false, b1,
                                                     (short)0, acc1, false, false);
    }

    #pragma unroll
    for (int r = 0; r < 8; ++r) {
        const int orow = mt + r + 8 * kh;
        y[(size_t)orow * PIX + p0 + nb + lm]      = acc0[r];
        y[(size_t)orow * PIX + p0 + nb + 16 + lm] = acc1[r];
    }
}

// ---------------------------------------------------------------------------
// Launch: norm A, build E, then y=b+E0@x; g1=A@x; y+=E1@g1; g2=A@g1; ...
// ws layout (floats): Anorm [512*512] | E [4*1024] | gA [SLABF] | gB [SLABF]
// ws bytes needed ~= 1MB + 16KB + 2*256MB ~= 514 MB.
// ---------------------------------------------------------------------------
extern "C" void kernel_launch(void* const* d_in, const int* in_sizes, int n_in,
                              void* d_out, int out_size, void* d_ws, size_t ws_size,
                              hipStream_t stream) {
    const float* x   = (const float*)d_in[0];
    const float* adj = (const float*)d_in[1];
    const float* W   = (const float*)d_in[2];
    const float* b   = (const float*)d_in[3];
    float* y  = (float*)d_out;
    float* ws = (float*)d_ws;

    float* Anorm = ws;
    float* E     = Anorm + (size_t)NV * NV;
    float* gA    = E + 4 * 1024;
    float* gB    = gA + SLABF;

    norm_adj_kernel<<<NV, 128, 0, stream>>>(adj, Anorm);
    make_eff_w_kernel<<<4, 256, 0, stream>>>(W, E);

    const dim3 convGrid(2048, NB);            // 2048 * 128 pixels = 512*512
    const dim3 diffGrid(8, 8, NB * CH);       // 64x64 tiles per 512x512 slab

    conv_kernel<<<convGrid, 256, 0, stream>>>(x, E + 0 * 1024, b, y, 1);
    diffuse_kernel<<<diffGrid, 256, 0, stream>>>(x, gA, Anorm);
    conv_kernel<<<convGrid, 256, 0, stream>>>(gA, E + 1 * 1024, b, y, 0);
    diffuse_kernel<<<diffGrid, 256, 0, stream>>>(gA, gB, Anorm);
    conv_kernel<<<convGrid, 256, 0, stream>>>(gB, E + 2 * 1024, b, y, 0);
    diffuse_kernel<<<diffGrid, 256, 0, stream>>>(gB, gA, Anorm);
    conv_kernel<<<convGrid, 256, 0, stream>>>(gA, E + 3 * 1024, b, y, 0);
}